// RGATLayer_12704513261598
// MI455X (gfx1250) — compile-verified
//
#include <hip/hip_runtime.h>
#include <hip/hip_bf16.h>

#define IN_DIM 128
#define DD 32

typedef __attribute__((ext_vector_type(2))) float v2f;
typedef __attribute__((ext_vector_type(8))) float v8f;

// Monotone float <-> int encoding so atomicMax(int) == float max.
__device__ __forceinline__ int enc_f32(float f) {
    int i = __float_as_int(f);
    return (i >= 0) ? i : (i ^ 0x7FFFFFFF);
}
__device__ __forceinline__ float dec_f32(int k) {
    int i = (k >= 0) ? k : (k ^ 0x7FFFFFFF);
    return __int_as_float(i);
}

// out = bias (accumulator base), m_enc = -inf encoding, denom = 0
__global__ void gat_init(float* __restrict__ out, int* __restrict__ m_enc,
                         float* __restrict__ denom, const float* __restrict__ bias, int N) {
    int i = blockIdx.x * blockDim.x + threadIdx.x;
    if (i < N * DD) out[i] = bias[i & (DD - 1)];
    if (i < N) { m_enc[i] = (int)0x80000000; denom[i] = 0.0f; }
}

// feat[N,32] = X[N,128] @ W[128,32] via V_WMMA_F32_16X16X4_F32.
// Also computes el[r] = feat[r]·attn_l, er[r] = feat[r]·attn_r via shfl reductions.
// One wave -> 16 rows x 32 cols (two 16x16 accumulators). 8 waves/block -> 128 rows.
__global__ __launch_bounds__(256) void gat_gemm_wmma(
    const float* __restrict__ X, const float* __restrict__ W,
    const float* __restrict__ attn_l, const float* __restrict__ attn_r,
    float* __restrict__ feat, float* __restrict__ el, float* __restrict__ er, int N)
{
    __shared__ float Wt[DD * IN_DIM];  // W transposed: Wt[n*128 + k] = W[k*32 + n]
    for (int idx = threadIdx.x; idx < DD * IN_DIM; idx += 256) {
        int n = idx & (DD - 1);
        int k = idx >> 5;
        Wt[n * IN_DIM + k] = W[k * DD + n];
    }
    __syncthreads();

    const int wave = threadIdx.x >> 5;
    const int lane = threadIdx.x & 31;
    const int half = lane >> 4;      // 0: K=0,1  1: K=2,3 (A layout, 16x4 f32)
    const int l    = lane & 15;

    const int row0 = blockIdx.x * 128 + wave * 16;
    int arow = row0 + l;
    if (arow >= N) arow = N - 1;     // clamp -> EXEC stays all-1s for WMMA
    const float* aptr = X + (long)arow * IN_DIM + 2 * half;

    v8f acc0 = {};  // cols 0..15
    v8f acc1 = {};  // cols 16..31

#pragma unroll 8
    for (int kk = 0; kk < IN_DIM / 4; ++kk) {
        const int k = kk * 4;
        v2f a  = *(const v2f*)(aptr + k);
        v2f b0 = *(const v2f*)(&Wt[l * IN_DIM + k + 2 * half]);
        v2f b1 = *(const v2f*)(&Wt[(l + 16) * IN_DIM + k + 2 * half]);
        acc0 = __builtin_amdgcn_wmma_f32_16x16x4_f32(false, a, false, b0,
                                                     (short)0, acc0, false, false);
        acc1 = __builtin_amdgcn_wmma_f32_16x16x4_f32(false, a, false, b1,
                                                     (short)0, acc1, false, false);
    }

    // attention vectors: this lane covers cols l and l+16
    const float al_lo = attn_l[l], al_hi = attn_l[l + 16];
    const float ar_lo = attn_r[l], ar_hi = attn_r[l + 16];

#pragma unroll
    for (int j = 0; j < 8; ++j) {
        const int r = row0 + j + 8 * half;  // C/D layout: VGPR j, half h -> row j+8h
        const float c0 = acc0[j];
        const float c1 = acc1[j];
        float pl = c0 * al_lo + c1 * al_hi;
        float pr = c0 * ar_lo + c1 * ar_hi;
#pragma unroll
        for (int m = 8; m >= 1; m >>= 1) {  // reduce within 16-lane half
            pl += __shfl_xor(pl, m, 32);
            pr += __shfl_xor(pr, m, 32);
        }
        if (r < N) {
            feat[(long)r * DD + l]      = c0;
            feat[(long)r * DD + 16 + l] = c1;
            if (l == 0) { el[r] = pl; er[r] = pr; }
        }
    }
}

// Pass A: per-edge logit + LeakyReLU, store e, segment-max into m_enc[dst].
__global__ void gat_edge_logits(const int* __restrict__ src, const int* __restrict__ dst,
                                const float* __restrict__ el, const float* __restrict__ er,
                                float* __restrict__ ebuf, int* __restrict__ m_enc, int E)
{
    int i = blockIdx.x * blockDim.x + threadIdx.x;
    if (i >= E) return;
    const int s = src[i], d = dst[i];
    float e = el[s] + er[d];
    e = (e > 0.0f) ? e : 0.2f * e;   // leaky_relu, NEG_SLOPE=0.2
    ebuf[i] = e;
    atomicMax(&m_enc[d], enc_f32(e));
}

// Pass B: ex = exp(e - m[dst]); segment-sum into denom[dst]; ebuf <- ex.
__global__ void gat_edge_exp(const int* __restrict__ dst, const int* __restrict__ m_enc,
                             float* __restrict__ ebuf, float* __restrict__ denom, int E)
{
    int i = blockIdx.x * blockDim.x + threadIdx.x;
    if (i >= E) return;
    const int d = dst[i];
    const float m = dec_f32(m_enc[d]);
    const float ex = __expf(ebuf[i] - m);
    ebuf[i] = ex;
    atomicAdd(&denom[d], ex);
}

// Pass C: wave per edge. Lane k handles feature dim k.
// Coalesced 128B gather of feat[src], 32 adjacent atomic f32 adds into out[dst].
__global__ __launch_bounds__(256) void gat_edge_aggregate(
    const int* __restrict__ src, const int* __restrict__ dst,
    const float* __restrict__ ebuf, const float* __restrict__ denom,
    const float* __restrict__ feat, float* __restrict__ out, int E)
{
    const long gid  = (long)blockIdx.x * blockDim.x + threadIdx.x;
    const long edge = gid >> 5;
    const int  lane = (int)(gid & 31);
    if (edge >= E) return;
    const int s = src[edge], d = dst[edge];
    const float den = denom[d];
    const float alpha = ebuf[edge] / ((den > 0.0f) ? den : 1.0f);
    const float v = feat[(long)s * DD + lane] * alpha;
    atomicAdd(&out[(long)d * DD + lane], v);
}

extern "C" void kernel_launch(void* const* d_in, const int* in_sizes, int n_in,
                              void* d_out, int out_size, void* d_ws, size_t ws_size,
                              hipStream_t stream) {
    const float* X      = (const float*)d_in[0];
    const int*   src    = (const int*)  d_in[1];
    const int*   dst    = (const int*)  d_in[2];
    const float* W      = (const float*)d_in[3];
    const float* attn_l = (const float*)d_in[4];
    const float* attn_r = (const float*)d_in[5];
    const float* bias   = (const float*)d_in[6];

    const int N = in_sizes[0] / IN_DIM;
    const int E = in_sizes[1];
    float* out = (float*)d_out;

    // Workspace carve-up (all fits comfortably in the 192MB L2):
    char* ws = (char*)d_ws;
    float* feat  = (float*)ws; ws += (size_t)N * DD * sizeof(float);  // 12.8 MB
    float* el    = (float*)ws; ws += (size_t)N * sizeof(float);
    float* er    = (float*)ws; ws += (size_t)N * sizeof(float);
    int*   m_enc = (int*)  ws; ws += (size_t)N * sizeof(int);
    float* denom = (float*)ws; ws += (size_t)N * sizeof(float);
    float* ebuf  = (float*)ws;                                         // 6.4 MB

    gat_init<<<(N * DD + 255) / 256, 256, 0, stream>>>(out, m_enc, denom, bias, N);

    gat_gemm_wmma<<<(N + 127) / 128, 256, 0, stream>>>(X, W, attn_l, attn_r,
                                                       feat, el, er, N);

    gat_edge_logits<<<(E + 255) / 256, 256, 0, stream>>>(src, dst, el, er,
                                                         ebuf, m_enc, E);

    gat_edge_exp<<<(E + 255) / 256, 256, 0, stream>>>(dst, m_enc, ebuf, denom, E);

    const int blocksC = (int)(((long)E * 32 + 255) / 256);
    gat_edge_aggregate<<<blocksC, 256, 0, stream>>>(src, dst, ebuf, denom,
                                                    feat, out, E);
}